// SAE_Hierarchical_40733469835425
// MI455X (gfx1250) — compile-verified
//
#include <hip/hip_runtime.h>
#include <hip/hip_bf16.h>

typedef __attribute__((ext_vector_type(2))) float v2f;
typedef __attribute__((ext_vector_type(8))) float v8f;

#define D_IN   64
#define HID    4096
#define HID2   2048
#define TOPK1  32
#define TOPK2  16
#define ROWS   16                 // rows of x per workgroup (one WMMA M-tile)
#define NWAVES 16                 // one wave per row for top-k phases
#define BLOCK  (NWAVES * 32)
#define XPAD   68                 // padded LDS row stride for 16x64 tiles (bank-conflict free)

// ---- LDS layout (floats) ----
#define OFF_XIN    0                          // 16 x 68   : x - b_dec
#define OFF_RC0    (OFF_XIN  + ROWS * XPAD)   // 16 x 68   : recon0
#define OFF_TOPV   (OFF_RC0  + ROWS * XPAD)   // 16 x 32   : top-k values
#define OFF_TOPI   (OFF_TOPV + ROWS * TOPK1)  // 16 x 32   : top-k indices (as int)
#define OFF_PRE    (OFF_TOPI + ROWS * TOPK1)  // 16 x 4096 : pre-activations (reused for level 1)
#define LDS_FLOATS (OFF_PRE  + ROWS * HID)    // = 68736 floats = 274944 bytes < 320KB/WGP

__global__ __launch_bounds__(BLOCK)
void sae_hier_fused_kernel(const float* __restrict__ x,
                           const float* __restrict__ W_enc,  const float* __restrict__ b_enc,
                           const float* __restrict__ W_dec,  const float* __restrict__ b_dec,
                           const float* __restrict__ W_enc2, const float* __restrict__ b_enc2,
                           const float* __restrict__ W_dec2, const float* __restrict__ b_dec2,
                           float* __restrict__ out)
{
    extern __shared__ float smem[];
    float* xin  = smem + OFF_XIN;
    float* rc0  = smem + OFF_RC0;
    float* topV = smem + OFF_TOPV;
    int*   topI = (int*)(smem + OFF_TOPI);
    float* pre  = smem + OFF_PRE;

    const int tid     = threadIdx.x;
    const int lane    = tid & 31;
    const int wave    = tid >> 5;
    const int rowBase = blockIdx.x * ROWS;

    // ---- stage 0: load x tile, subtract b_dec ----
    for (int i = tid; i < ROWS * D_IN; i += BLOCK) {
        const int r = i >> 6, c = i & 63;
        xin[r * XPAD + c] = x[(size_t)(rowBase + r) * D_IN + c] - b_dec[c];
    }
    __syncthreads();

    // WMMA f32 16x16x4 A-fragment lane mapping: M = lane%16, K-pair base = lane<16 ? 0 : 2
    const int mrow = lane & 15;
    const int kb   = (lane < 16) ? 0 : 2;
    const int Mofs = (lane < 16) ? 0 : 8;     // C/D: VGPR r holds M = r + Mofs, N = lane%16

    // ---- level 0 GEMM: pre = relu(xin @ W_enc^T + b_enc), N split across 16 waves ----
    {
        v2f afrag[16];
        #pragma unroll
        for (int kk = 0; kk < 16; ++kk) {
            afrag[kk].x = xin[mrow * XPAD + 4 * kk + kb];
            afrag[kk].y = xin[mrow * XPAD + 4 * kk + kb + 1];
        }
        for (int chunk = wave; chunk < HID / 16; chunk += NWAVES) {
            const int n0 = chunk * 16;
            const float* wr = W_enc + (size_t)(n0 + mrow) * D_IN;   // B(k,n) = W_enc[n][k]
            v8f c = {};
            #pragma unroll
            for (int kk = 0; kk < 16; ++kk) {
                v2f b = *(const v2f*)(wr + 4 * kk + kb);            // global_load_b64, L2-hit
                c = __builtin_amdgcn_wmma_f32_16x16x4_f32(false, afrag[kk], false, b,
                                                          (short)0, c, false, false);
            }
            const float bias = b_enc[n0 + mrow];
            #pragma unroll
            for (int r = 0; r < 8; ++r) {
                float v = c[r] + bias;
                v = v > 0.f ? v : 0.f;                               // relu
                pre[(r + Mofs) * HID + n0 + mrow] = v;
            }
        }
    }
    __syncthreads();

    // ---- level 0 top-32 + sparse decode: wave w owns row w ----
    {
        const int row = wave;
        float* prow = pre + row * HID;
        for (int t = 0; t < TOPK1; ++t) {
            float best = -__builtin_inff(); int bidx = 0x7fffffff;
            for (int j = 0; j < HID / 32; ++j) {                     // interleaved strips: no bank conflicts
                const int idx = j * 32 + lane;
                const float v = prow[idx];
                if (v > best) { best = v; bidx = idx; }
            }
            #pragma unroll
            for (int off = 16; off > 0; off >>= 1) {                 // wave32 xor-argmax (all lanes converge)
                const float ov = __shfl_xor(best, off, 32);
                const int   oi = __shfl_xor(bidx, off, 32);
                if (ov > best || (ov == best && oi < bidx)) { best = ov; bidx = oi; }
            }
            if (lane == 0) { topV[row * TOPK1 + t] = best; topI[row * TOPK1 + t] = bidx; }
            if ((bidx & 31) == lane) prow[bidx] = -__builtin_inff(); // extract
        }
        // recon0[row] = b_dec + sum v_t * W_dec[idx_t]  (lane handles cols lane, lane+32)
        float a0 = b_dec[lane], a1 = b_dec[lane + 32];
        for (int t = 0; t < TOPK1; ++t) {
            const float v  = topV[row * TOPK1 + t];
            const int   ix = topI[row * TOPK1 + t];
            const float* dr = W_dec + (size_t)ix * D_IN;
            a0 += v * dr[lane];
            a1 += v * dr[lane + 32];
        }
        rc0[row * XPAD + lane]      = a0;
        rc0[row * XPAD + lane + 32] = a1;
    }
    __syncthreads();

    // ---- level 1 GEMM: pre2 = relu(recon0 @ W_enc2^T + b_enc2) ----
    {
        v2f afrag[16];
        #pragma unroll
        for (int kk = 0; kk < 16; ++kk) {
            afrag[kk].x = rc0[mrow * XPAD + 4 * kk + kb];
            afrag[kk].y = rc0[mrow * XPAD + 4 * kk + kb + 1];
        }
        for (int chunk = wave; chunk < HID2 / 16; chunk += NWAVES) {
            const int n0 = chunk * 16;
            const float* wr = W_enc2 + (size_t)(n0 + mrow) * D_IN;
            v8f c = {};
            #pragma unroll
            for (int kk = 0; kk < 16; ++kk) {
                v2f b = *(const v2f*)(wr + 4 * kk + kb);
                c = __builtin_amdgcn_wmma_f32_16x16x4_f32(false, afrag[kk], false, b,
                                                          (short)0, c, false, false);
            }
            const float bias = b_enc2[n0 + mrow];
            #pragma unroll
            for (int r = 0; r < 8; ++r) {
                float v = c[r] + bias;
                pre[(r + Mofs) * HID2 + n0 + mrow] = v > 0.f ? v : 0.f;
            }
        }
    }
    __syncthreads();

    // ---- level 1 top-16, sparse decode, decay-weighted combine, store ----
    {
        const int row = wave;
        float* prow = pre + row * HID2;
        for (int t = 0; t < TOPK2; ++t) {
            float best = -__builtin_inff(); int bidx = 0x7fffffff;
            for (int j = 0; j < HID2 / 32; ++j) {
                const int idx = j * 32 + lane;
                const float v = prow[idx];
                if (v > best) { best = v; bidx = idx; }
            }
            #pragma unroll
            for (int off = 16; off > 0; off >>= 1) {
                const float ov = __shfl_xor(best, off, 32);
                const int   oi = __shfl_xor(bidx, off, 32);
                if (ov > best || (ov == best && oi < bidx)) { best = ov; bidx = oi; }
            }
            if (lane == 0) { topV[row * TOPK1 + t] = best; topI[row * TOPK1 + t] = bidx; }
            if ((bidx & 31) == lane) prow[bidx] = -__builtin_inff();
        }
        float a0 = b_dec2[lane], a1 = b_dec2[lane + 32];
        for (int t = 0; t < TOPK2; ++t) {
            const float v  = topV[row * TOPK1 + t];
            const int   ix = topI[row * TOPK1 + t];
            const float* dr = W_dec2 + (size_t)ix * D_IN;
            a0 += v * dr[lane];
            a1 += v * dr[lane + 32];
        }
        const float w0 = 1.0f / 1.5f, w1 = 0.5f / 1.5f;   // [1, DECAY]/sum
        const size_t base = (size_t)(rowBase + row) * D_IN;
        out[base + lane]      = w0 * rc0[row * XPAD + lane]      + w1 * a0;
        out[base + lane + 32] = w0 * rc0[row * XPAD + lane + 32] + w1 * a1;
    }
}

extern "C" void kernel_launch(void* const* d_in, const int* in_sizes, int n_in,
                              void* d_out, int out_size, void* d_ws, size_t ws_size,
                              hipStream_t stream) {
    (void)n_in; (void)d_ws; (void)ws_size; (void)out_size;
    const float* x      = (const float*)d_in[0];
    const float* W_enc  = (const float*)d_in[1];
    const float* b_enc  = (const float*)d_in[2];
    const float* W_dec  = (const float*)d_in[3];
    const float* b_dec  = (const float*)d_in[4];
    const float* W_enc2 = (const float*)d_in[5];
    const float* b_enc2 = (const float*)d_in[6];
    const float* W_dec2 = (const float*)d_in[7];
    const float* b_dec2 = (const float*)d_in[8];
    float* out = (float*)d_out;

    const int batch  = in_sizes[0] / D_IN;       // 16384
    const int nblk   = batch / ROWS;             // 1024 workgroups (1 per WGP resident)
    const size_t lds = (size_t)LDS_FLOATS * sizeof(float);   // 274944 B < 320 KB/WGP

    sae_hier_fused_kernel<<<nblk, BLOCK, lds, stream>>>(
        x, W_enc, b_enc, W_dec, b_dec, W_enc2, b_enc2, W_dec2, b_dec2, out);
}